// SimpleModel_14551349199008
// MI455X (gfx1250) — compile-verified
//
#include <hip/hip_runtime.h>

#define NN   100000
#define EE   1600000
#define KIN  128
#define HH   64
#define LL   3
#define ET   (EE + NN)
#define GRAM_CHUNK 4000

typedef float v2f __attribute__((ext_vector_type(2)));
typedef float v8f __attribute__((ext_vector_type(8)));

// ---------------- utility kernels ----------------
__global__ void zero_kernel(float* p, long long n) {
  long long t = (long long)blockIdx.x * blockDim.x + threadIdx.x;
  long long stride = (long long)gridDim.x * blockDim.x;
  for (; t < n; t += stride) p[t] = 0.f;
}

__global__ void copy4_kernel(const float* __restrict__ src, float* __restrict__ dst, long long n4) {
  long long t = (long long)blockIdx.x * blockDim.x + threadIdx.x;
  long long stride = (long long)gridDim.x * blockDim.x;
  for (; t < n4; t += stride) ((float4*)dst)[t] = ((const float4*)src)[t];
}

// ---------------- graph preprocessing ----------------
__global__ void deg_kernel(const int* __restrict__ srcE, const int* __restrict__ dstE,
                           float* degSrc, float* degDst) {
  int e = blockIdx.x * blockDim.x + threadIdx.x;
  if (e >= ET) return;
  int s, d;
  if (e < EE) { s = srcE[e]; d = dstE[e]; } else { s = d = e - EE; }
  atomicAdd(&degSrc[s], 1.f);
  atomicAdd(&degDst[d], 1.f);
}

__global__ void rsqrt_kernel(float* p, int n) {
  int i = blockIdx.x * blockDim.x + threadIdx.x;
  if (i < n) p[i] = rsqrtf(p[i]);
}

__global__ void norm_kernel(const float* __restrict__ dis, const int* __restrict__ srcE,
                            const int* __restrict__ dstE, float* __restrict__ nrm) {
  int e = blockIdx.x * blockDim.x + threadIdx.x;
  if (e >= ET) return;
  int s, d;
  if (e < EE) { s = srcE[e]; d = dstE[e]; } else { s = d = e - EE; }
  nrm[e] = dis[s] * dis[d];
}

// ---------------- WMMA fp32 GEMM:  C[nrows,64] = A[nrows,K] @ W[K,64] (+bias) ----------------
// Each wave produces one 16x64 output tile using 4 accumulators of V_WMMA_F32_16X16X4_F32.
__global__ void gemm_wmma_kernel(const float* __restrict__ A, const float* __restrict__ W,
                                 const float* __restrict__ bias, float* __restrict__ C,
                                 int nrows, int K) {
  const int lane = threadIdx.x & 31;
  const int wave = blockIdx.x * (blockDim.x >> 5) + (threadIdx.x >> 5);
  const int row0 = wave * 16;
  if (row0 >= nrows) return;              // wave-uniform: EXEC stays all-ones for WMMA
  const int ml = lane & 15;
  const int kh = (lane >> 4) << 1;        // 0 for lanes 0-15 (K 0,1), 2 for lanes 16-31 (K 2,3)
  v8f c0 = {}, c1 = {}, c2 = {}, c3 = {};
  for (int k = 0; k < K; k += 4) {
    const float* ap = A + (size_t)(row0 + ml) * K + (k + kh);
    v2f a; a.x = ap[0]; a.y = ap[1];
    const float* wp = W + (size_t)(k + kh) * 64 + ml;   // B VGPR0: rows {k, k+2}; VGPR1: rows {k+1, k+3}
    v2f b0, b1, b2, b3;
    b0.x = wp[0];  b0.y = wp[64];
    b1.x = wp[16]; b1.y = wp[64 + 16];
    b2.x = wp[32]; b2.y = wp[64 + 32];
    b3.x = wp[48]; b3.y = wp[64 + 48];
    c0 = __builtin_amdgcn_wmma_f32_16x16x4_f32(false, a, false, b0, (short)0, c0, false, false);
    c1 = __builtin_amdgcn_wmma_f32_16x16x4_f32(false, a, false, b1, (short)0, c1, false, false);
    c2 = __builtin_amdgcn_wmma_f32_16x16x4_f32(false, a, false, b2, (short)0, c2, false, false);
    c3 = __builtin_amdgcn_wmma_f32_16x16x4_f32(false, a, false, b3, (short)0, c3, false, false);
  }
  float bv0 = 0.f, bv1 = 0.f, bv2 = 0.f, bv3 = 0.f;
  if (bias) { bv0 = bias[ml]; bv1 = bias[16 + ml]; bv2 = bias[32 + ml]; bv3 = bias[48 + ml]; }
  const int rb = row0 + ((lane >> 4) << 3);   // C VGPR v: rows v (lanes 0-15) / v+8 (lanes 16-31)
  for (int v = 0; v < 8; ++v) {
    float* cp = C + (size_t)(rb + v) * 64 + ml;
    cp[0]  = c0[v] + bv0;
    cp[16] = c1[v] + bv1;
    cp[32] = c2[v] + bv2;
    cp[48] = c3[v] + bv3;
  }
}

// ---------------- WMMA Gram:  G[64,64] += H^T H over a node chunk ----------------
// Block = 512 threads = 16 waves; wave w owns 16x16 tile (mi, ni) of G.
__global__ void gram_wmma_kernel(const float* __restrict__ H, int nrows, float* __restrict__ G) {
  const int lane = threadIdx.x & 31;
  const int w = threadIdx.x >> 5;
  const int mi = (w >> 2) << 4;
  const int ni = (w & 3) << 4;
  const int ml = lane & 15;
  const int kh = (lane >> 4) << 1;
  int k0 = blockIdx.x * GRAM_CHUNK;
  int k1 = k0 + GRAM_CHUNK; if (k1 > nrows) k1 = nrows;
  v8f acc = {};
  for (int k = k0; k < k1; k += 4) {
    const float* r0 = H + (size_t)(k + kh) * 64;      // A[m][kk] = H[kk][m]
    const float* r1 = H + (size_t)(k + kh + 1) * 64;
    v2f a; a.x = r0[mi + ml]; a.y = r1[mi + ml];
    v2f b; b.x = r0[ni + ml]; b.y = r1[ni + ml];      // B[kk][n] = H[kk][n]
    acc = __builtin_amdgcn_wmma_f32_16x16x4_f32(false, a, false, b, (short)0, acc, false, false);
  }
  const int rb = mi + ((lane >> 4) << 3);
  for (int v = 0; v < 8; ++v)
    atomicAdd(&G[(rb + v) * 64 + ni + ml], acc[v]);
}

// ---------------- message passing ----------------
// 16 threads per edge, 4 channels (float4) each: h[dst] += xw[src] * norm
__global__ void scatter_kernel(const float* __restrict__ xw, const float* __restrict__ nrm,
                               const int* __restrict__ srcE, const int* __restrict__ dstE,
                               float* __restrict__ out) {
  long long t = (long long)blockIdx.x * blockDim.x + threadIdx.x;
  long long e = t >> 4;
  if (e >= ET) return;
  int c4 = ((int)t & 15) * 4;
  int s, d;
  if (e < EE) { s = srcE[e]; d = dstE[e]; } else { s = d = (int)(e - EE); }
  float w = nrm[e];
  float4 v = *(const float4*)(xw + (size_t)s * 64 + c4);
  float* o = out + (size_t)d * 64 + c4;
  atomicAdd(o + 0, v.x * w);
  atomicAdd(o + 1, v.y * w);
  atomicAdd(o + 2, v.z * w);
  atomicAdd(o + 3, v.w * w);
}

__global__ void bias_relu_kernel(float* __restrict__ h, const float* __restrict__ bias) {
  long long t = (long long)blockIdx.x * blockDim.x + threadIdx.x;
  long long stride = (long long)gridDim.x * blockDim.x;
  long long n = (long long)NN * 64;
  for (; t < n; t += stride) {
    int c = (int)(t & 63);
    h[t] = fmaxf(h[t] + bias[c], 0.f);
  }
}

// ---------------- reductions for Dirichlet energies ----------------
__global__ void frob_kernel(const float* __restrict__ H, long long n4, float* accum) {
  long long t = (long long)blockIdx.x * blockDim.x + threadIdx.x;
  float local = 0.f;
  if (t < n4) {
    float4 v = ((const float4*)H)[t];
    local = v.x * v.x + v.y * v.y + v.z * v.z + v.w * v.w;
  }
  for (int off = 16; off > 0; off >>= 1) local += __shfl_down(local, off, 32);
  if ((threadIdx.x & 31) == 0) atomicAdd(accum, local);
}

__global__ void dir_rw_kernel(const float* __restrict__ H, const int* __restrict__ srcE,
                              const int* __restrict__ dstE, float* accum) {
  int e = blockIdx.x * blockDim.x + threadIdx.x;
  float local = 0.f;
  if (e < ET) {
    int s, d;
    if (e < EE) { s = srcE[e]; d = dstE[e]; } else { s = d = e - EE; }
    if (s != d) {
      const float4* rs = (const float4*)(H + (size_t)s * 64);
      const float4* rd = (const float4*)(H + (size_t)d * 64);
      for (int i = 0; i < 16; ++i) {
        float4 a = rs[i], b = rd[i];
        float dx = a.x - b.x, dy = a.y - b.y, dz = a.z - b.z, dw = a.w - b.w;
        local += dx * dx + dy * dy + dz * dz + dw * dw;
      }
    }
  }
  for (int off = 16; off > 0; off >>= 1) local += __shfl_down(local, off, 32);
  if ((threadIdx.x & 31) == 0) atomicAdd(accum, local);
}

__global__ void dir_sym_kernel(const float* __restrict__ H, const float* __restrict__ degSrc,
                               const int* __restrict__ srcE, const int* __restrict__ dstE,
                               float* accum) {
  int e = blockIdx.x * blockDim.x + threadIdx.x;
  float local = 0.f;
  if (e < ET) {
    int s, d;
    if (e < EE) { s = srcE[e]; d = dstE[e]; } else { s = d = e - EE; }
    if (s != d) {
      float ds_ = rsqrtf(degSrc[s]);
      float dd_ = rsqrtf(degSrc[d]);
      float coef = ds_ * dd_;                       // rsqrt(deg[s]*deg[d])
      const float4* rs = (const float4*)(H + (size_t)s * 64);
      const float4* rd = (const float4*)(H + (size_t)d * 64);
      float acc = 0.f;
      for (int i = 0; i < 16; ++i) {
        float4 a = rs[i], b = rd[i];
        float dx = a.x * ds_ - b.x * dd_;
        float dy = a.y * ds_ - b.y * dd_;
        float dz = a.z * ds_ - b.z * dd_;
        float dw = a.w * ds_ - b.w * dd_;
        acc += dx * dx + dy * dy + dz * dz + dw * dw;
      }
      local = coef * acc;
    }
  }
  for (int off = 16; off > 0; off >>= 1) local += __shfl_down(local, off, 32);
  if ((threadIdx.x & 31) == 0) atomicAdd(accum, local);
}

__global__ void write_energy_kernel(float* dst, const float* num, const float* den) {
  if (threadIdx.x == 0 && blockIdx.x == 0) *dst = 0.5f * (*num) / (*den);
}

// ---------------- rank-diff machinery ----------------
__global__ void rowsum_argmax_kernel(const float* __restrict__ H, unsigned long long* key) {
  int node = blockIdx.x * blockDim.x + threadIdx.x;
  if (node >= NN) return;
  const float4* r = (const float4*)(H + (size_t)node * 64);
  float ssum = 0.f;
  for (int i = 0; i < 16; ++i) {
    float4 v = r[i];
    ssum += fabsf(v.x) + fabsf(v.y) + fabsf(v.z) + fabsf(v.w);
  }
  unsigned long long k = (((unsigned long long)__float_as_uint(ssum)) << 32) | (unsigned)node;
  atomicMax(key, k);
}

__global__ void colsum_kernel(const float* __restrict__ H, float* colsum) {
  int tid = blockIdx.x * blockDim.x + threadIdx.x;
  int c = tid & 63;
  int grp = tid >> 6;
  int ngrp = (gridDim.x * blockDim.x) >> 6;
  float local = 0.f;
  for (int node = grp; node < NN; node += ngrp)
    local += fabsf(H[(size_t)node * 64 + c]);
  atomicAdd(&colsum[c], local);
}

__global__ void colargmax_kernel(const float* __restrict__ cs, int* j) {
  if (threadIdx.x == 0 && blockIdx.x == 0) {
    float best = cs[0]; int bj = 0;
    for (int c = 1; c < 64; ++c) if (cs[c] > best) { best = cs[c]; bj = c; }
    *j = bj;
  }
}

// G2 = Gn - beta*(u m0^T + m0 u^T) + m0 m0^T / ||m0||^2  (analytic rank-1 deflation)
__global__ void build_g2_kernel(const float* __restrict__ G, const float* __restrict__ H,
                                const unsigned long long* keyPtr, const int* jPtr,
                                const float* nucPtr, float* __restrict__ G2) {
  __shared__ float m0[64], u[64], par[3];
  const int t = threadIdx.x;
  const int i = (int)((*keyPtr) & 0xffffffffull);
  const int j = *jPtr;
  const float nuc = *nucPtr;
  const float inv2 = 1.f / (nuc * nuc);
  if (t < 64) {
    m0[t] = H[(size_t)i * 64 + t];
    u[t] = G[t * 64 + j] * inv2;
  }
  __syncthreads();
  if (t == 0) {
    float sgn = (m0[j] < 0.f) ? -1.f : 1.f;
    float ssum = 0.f;
    for (int c = 0; c < 64; ++c) ssum += m0[c] * m0[c];
    float m0sq = ssum * inv2;
    float m1sq = G[j * 64 + j] * inv2;
    par[0] = rsqrtf(fmaxf(m1sq * m0sq, 1e-30f));   // beta = 1/(||m1|| ||m0||)
    par[1] = 1.f / fmaxf(m0sq, 1e-30f);
    par[2] = sgn / nuc;
  }
  __syncthreads();
  if (t < 64) m0[t] *= par[2];
  __syncthreads();
  const float beta = par[0], invm0 = par[1];
  for (int idx = t; idx < 4096; idx += blockDim.x) {
    int a = idx >> 6, b = idx & 63;
    G2[idx] = G[idx] * inv2 - beta * (u[a] * m0[b] + m0[a] * u[b]) + m0[a] * m0[b] * invm0;
  }
}

// Cyclic Jacobi eigensolver, 64x64 symmetric, one workgroup of 64 threads (2 waves).
// Writes sum(sqrt(clip(eigvals,0))) = nuclear norm.
__global__ void jacobi64_kernel(const float* __restrict__ Gin, float* __restrict__ out) {
  __shared__ float A[64][65];
  __shared__ float cs[2];
  __shared__ float red[64];
  const int t = threadIdx.x;
  for (int r = 0; r < 64; ++r) A[r][t] = Gin[r * 64 + t];
  __syncthreads();
  for (int sweep = 0; sweep < 10; ++sweep) {
    for (int p = 0; p < 63; ++p) {
      for (int q = p + 1; q < 64; ++q) {
        if (t == 0) {
          float apq = A[p][q];
          float c = 1.f, s = 0.f;
          if (fabsf(apq) > 1e-12f) {
            float tau = (A[q][q] - A[p][p]) / (2.f * apq);
            float tt = (tau >= 0.f ? 1.f : -1.f) / (fabsf(tau) + sqrtf(1.f + tau * tau));
            c = rsqrtf(1.f + tt * tt);
            s = tt * c;
          }
          cs[0] = c; cs[1] = s;
        }
        __syncthreads();
        float c = cs[0], s = cs[1];        // block-uniform
        if (s != 0.f) {
          float akp = A[t][p], akq = A[t][q];
          A[t][p] = c * akp - s * akq;
          A[t][q] = s * akp + c * akq;
          __syncthreads();
          float apk = A[p][t], aqk = A[q][t];
          A[p][t] = c * apk - s * aqk;
          A[q][t] = s * apk + c * aqk;
        }
        __syncthreads();
      }
    }
  }
  red[t] = sqrtf(fmaxf(A[t][t], 0.f));
  __syncthreads();
  for (int off = 32; off > 0; off >>= 1) {
    if (t < off) red[t] += red[t + off];
    __syncthreads();
  }
  if (t == 0) *out = red[0];
}

// ---------------- host-side pipeline ----------------
extern "C" void kernel_launch(void* const* d_in, const int* in_sizes, int n_in,
                              void* d_out, int out_size, void* d_ws, size_t ws_size,
                              hipStream_t stream) {
  (void)in_sizes; (void)n_in; (void)out_size; (void)ws_size;
  const float* x     = (const float*)d_in[0];
  const int*   ei    = (const int*)d_in[1];
  const float* encw  = (const float*)d_in[2];
  const float* encb  = (const float*)d_in[3];
  const float* convw = (const float*)d_in[4];
  const float* convb = (const float*)d_in[5];
  float* out = (float*)d_out;
  const int* srcE = ei;
  const int* dstE = ei + EE;

  float* ws     = (float*)d_ws;
  float* degSrc = ws;                                 // N
  float* dis    = ws + (size_t)NN;                    // N  (rsqrt of deg_dst)
  float* normw  = ws + (size_t)2 * NN;                // ET
  float* hbuf   = normw + (size_t)ET;                 // N*64
  float* xwbuf  = hbuf + (size_t)NN * HH;             // N*64
  float* G      = xwbuf + (size_t)NN * HH;            // 4096
  float* G2     = G + 64 * 64;                        // 4096
  float* scal   = G2 + 64 * 64;                       // [0]=frob2 [1]=rw [2]=sym [3]=nuc1
  unsigned long long* key = (unsigned long long*)(scal + 16);   // 8-byte aligned
  float* colsum = scal + 18;                          // 64
  int* jidx     = (int*)(colsum + 64);

  const int T = 256;
  const int ETB = (ET + T - 1) / T;
  const int GEMMB = ((NN / 16) + 7) / 8;
  const long long OUTH = (long long)NN * HH;          // 6400000

  // degrees + GCN norm
  zero_kernel<<<1024, T, 0, stream>>>(degSrc, (long long)2 * NN);
  deg_kernel<<<ETB, T, 0, stream>>>(srcE, dstE, degSrc, dis);
  rsqrt_kernel<<<(NN + T - 1) / T, T, 0, stream>>>(dis, NN);
  norm_kernel<<<ETB, T, 0, stream>>>(dis, srcE, dstE, normw);

  // encoder: h = x @ enc_w + enc_b   (WMMA fp32)
  gemm_wmma_kernel<<<GEMMB, T, 0, stream>>>(x, encw, encb, hbuf, NN, KIN);
  zero_kernel<<<1, 32, 0, stream>>>(scal, 16);
  frob_kernel<<<(int)((OUTH / 4 + T - 1) / T), T, 0, stream>>>(hbuf, OUTH / 4, scal + 0);
  dir_sym_kernel<<<ETB, T, 0, stream>>>(hbuf, degSrc, srcE, dstE, scal + 2);
  write_energy_kernel<<<1, 32, 0, stream>>>(out + OUTH + 0, scal + 2, scal + 0);

  for (int l = 0; l < LL; ++l) {
    // GCNConv
    gemm_wmma_kernel<<<GEMMB, T, 0, stream>>>(hbuf, convw + (size_t)l * HH * HH, nullptr, xwbuf, NN, HH);
    zero_kernel<<<4096, T, 0, stream>>>(hbuf, OUTH);
    scatter_kernel<<<(int)(((long long)ET * 16 + T - 1) / T), T, 0, stream>>>(xwbuf, normw, srcE, dstE, hbuf);
    bias_relu_kernel<<<4096, T, 0, stream>>>(hbuf, convb + (size_t)l * HH);

    // Dirichlet energies
    zero_kernel<<<1, 32, 0, stream>>>(scal, 16);
    frob_kernel<<<(int)((OUTH / 4 + T - 1) / T), T, 0, stream>>>(hbuf, OUTH / 4, scal + 0);
    dir_rw_kernel<<<ETB, T, 0, stream>>>(hbuf, srcE, dstE, scal + 1);
    dir_sym_kernel<<<ETB, T, 0, stream>>>(hbuf, degSrc, srcE, dstE, scal + 2);
    write_energy_kernel<<<1, 32, 0, stream>>>(out + OUTH + 1 + l, scal + 1, scal + 0);
    write_energy_kernel<<<1, 32, 0, stream>>>(out + OUTH + 4 + l, scal + 2, scal + 0);

    // rank diff: Gram (WMMA) -> Jacobi -> rank-1 deflation -> Jacobi
    zero_kernel<<<16, T, 0, stream>>>(G, 64 * 64);
    zero_kernel<<<1, 128, 0, stream>>>(scal + 16, 2 + 64 + 1);   // key, colsum, jidx
    gram_wmma_kernel<<<(NN + GRAM_CHUNK - 1) / GRAM_CHUNK, 512, 0, stream>>>(hbuf, NN, G);
    jacobi64_kernel<<<1, 64, 0, stream>>>(G, scal + 3);          // nuc1
    rowsum_argmax_kernel<<<(NN + T - 1) / T, T, 0, stream>>>(hbuf, key);
    colsum_kernel<<<128, T, 0, stream>>>(hbuf, colsum);
    colargmax_kernel<<<1, 32, 0, stream>>>(colsum, jidx);
    build_g2_kernel<<<1, 256, 0, stream>>>(G, hbuf, key, jidx, scal + 3, G2);
    jacobi64_kernel<<<1, 64, 0, stream>>>(G2, out + OUTH + 7 + l);
  }

  copy4_kernel<<<4096, T, 0, stream>>>(hbuf, out, OUTH / 4);
}